// Network_61658550501609
// MI455X (gfx1250) — compile-verified
//
#include <hip/hip_runtime.h>
#include <hip/hip_bf16.h>
#include <math.h>
#include <stdint.h>

// ---------------- problem constants ----------------
#define BB 32
#define LL 8192
#define DM 41          // d_model
#define DSTATE 16
#define DCONV 4
#define DI 82          // d_inner
#define DTR 3          // dt_rank
#define NL 10          // num labels
#define E2 164         // 2*DI
#define XP 35          // DTR + 2*DSTATE
#define MTOT (BB*LL)   // 262144 rows
#define REC 196        // per-timestep record: delta(82) xc(82) B(16) C(16)
#define KXP 44         // x K padded 41 -> 44
#define XZS 176        // xz row stride (164 -> 176, last WMMA N-tile stores unguarded)
#define YS  84         // yg row stride (82 -> 84, zero pads -> unguarded logits K loop)
#define NC  32         // scan chunks per batch
#define CL  256        // chunk length (LL/NC)
#define TT  16         // timesteps staged per LDS tile

typedef float v2f __attribute__((ext_vector_type(2)));
typedef float v8f __attribute__((ext_vector_type(8)));

__device__ __forceinline__ float silu_f(float x) {
  return x / (1.0f + __expf(-x));
}
__device__ __forceinline__ float softplus_f(float x) {
  return (x > 20.0f) ? x : __logf(1.0f + __expf(x));
}

// Issue one per-lane async 16B global->LDS transfer (gfx1250), EXEC-masked.
__device__ __forceinline__ void async_g2l_b128(const float* g, float* lds) {
  unsigned long long ga = (unsigned long long)(uintptr_t)g;
  unsigned lo = (unsigned)(uintptr_t)lds;   // low 32 bits of generic = LDS offset
  asm volatile("global_load_async_to_lds_b128 %0, %1, off"
               :: "v"(lo), "v"(ga) : "memory");
}
__device__ __forceinline__ void async_wait0() {
  asm volatile("s_wait_asynccnt 0x0" ::: "memory");
}

// ---------------- prep: pad x to (M x 44), zeros at k=41..43 ----------------
__global__ void kp_xpad(const float* __restrict__ x, float* __restrict__ xP) {
  long i = (long)blockIdx.x * blockDim.x + threadIdx.x;
  if (i >= (long)MTOT * KXP) return;
  long m = i / KXP;
  int  k = (int)(i - m * KXP);
  xP[i] = (k < DM) ? x[m * DM + k] : 0.f;
}

// ---------------- prep: in_proj_w -> WMMA B-fragment image ----------------
// WinP[((ntile*22 + kpair)*16 + n16)*2 + r] = W^T[k=2*kpair+r][n=ntile*16+n16]
__global__ void kp_win(const float* __restrict__ w, float* __restrict__ WinP) {
  for (int idx = threadIdx.x; idx < 11 * 22 * 16 * 2; idx += blockDim.x) {
    int r = idx & 1, n16 = (idx >> 1) & 15;
    int kp = (idx >> 5) % 22, nt = (idx >> 5) / 22;
    int k = kp * 2 + r, n = nt * 16 + n16;
    WinP[idx] = (k < DM && n < E2) ? w[n * DM + k] : 0.f;
  }
}

// ---------------- prep: Wc = cls_w @ out_proj_w in B-fragment order --------
// WcP2[(kpair*16 + n16)*2 + r] = Wc[n16][k=2*kpair+r], K padded 82->84
__global__ void k0_wc(const float* __restrict__ cls_w,
                      const float* __restrict__ out_proj_w,
                      float* __restrict__ WcP2) {
  for (int idx = threadIdx.x; idx < 42 * 16 * 2; idx += blockDim.x) {
    int r = idx & 1, n = (idx >> 1) & 15, kp = idx >> 5;
    int k = kp * 2 + r;                       // inner-channel index d
    float acc = 0.f;
    if (k < DI && n < NL) {
      for (int m = 0; m < DM; ++m)
        acc = fmaf(cls_w[n * DM + m], out_proj_w[m * DI + k], acc);
    }
    WcP2[idx] = acc;
  }
}

// ---------------- k1: xz = x (M x 41) @ in_proj_w^T via fp32 WMMA ----------
// One wave per 16-row M-tile; 11 A-fragments held in registers, reused for
// all 11 N-tiles. All operand loads are unconditional aligned float2.
__global__ void k1_inproj(const float* __restrict__ xP,
                          const float* __restrict__ WinP,
                          float* __restrict__ xz) {
  const int lane = threadIdx.x & 31;
  const int wid  = blockIdx.x * (blockDim.x >> 5) + (threadIdx.x >> 5);
  if (wid >= MTOT / 16) return;
  const int l16 = lane & 15, hi = lane >> 4;
  const long m_a = (long)wid * 16 + l16;
  const float* ap = xP + m_a * KXP + hi * 2;
  v2f a[11];
#pragma unroll
  for (int ks = 0; ks < 11; ++ks) a[ks] = *(const v2f*)(ap + ks * 4);

  for (int nt = 0; nt < 11; ++nt) {
    v8f c = {};
#pragma unroll
    for (int ks = 0; ks < 11; ++ks) {
      const int kp = ks * 2 + hi;
      v2f bv = *(const v2f*)(WinP + ((nt * 22 + kp) * 16 + l16) * 2);
      c = __builtin_amdgcn_wmma_f32_16x16x4_f32(false, a[ks], false, bv,
                                                (short)0, c, false, false);
    }
#pragma unroll
    for (int r = 0; r < 8; ++r) {
      const long m = (long)wid * 16 + r + hi * 8;
      xz[m * XZS + nt * 16 + l16] = c[r];
    }
  }
}

// ---------------- k2: conv(4)+SiLU, x_proj, dt_proj+softplus -> record ------
#define TL 32
__global__ void k2_feat(const float* __restrict__ xz,
                        const float* __restrict__ conv_w,
                        const float* __restrict__ conv_b,
                        const float* __restrict__ x_proj_w,
                        const float* __restrict__ dt_proj_w,
                        const float* __restrict__ dt_proj_b,
                        float* __restrict__ rec) {
  __shared__ float xsT[(TL + 3) * 84];
  __shared__ float xcT[TL * 84];
  __shared__ float xdT[TL * 36];
  const int b   = blockIdx.x >> 8;          // LL/TL = 256 tiles per batch
  const int l0  = (blockIdx.x & 255) * TL;
  const int tid = threadIdx.x;

  for (int idx = tid; idx < (TL + 3) * DI; idx += blockDim.x) {
    int j = idx / DI, d = idx % DI;
    int l = l0 - 3 + j;                     // causal halo, zero pad at l<0
    xsT[j * 84 + d] = (l >= 0) ? xz[((long)b * LL + l) * XZS + d] : 0.f;
  }
  __syncthreads();

  for (int idx = tid; idx < TL * DI; idx += blockDim.x) {
    int t = idx / DI, d = idx % DI;
    float acc = conv_b[d];
#pragma unroll
    for (int k = 0; k < DCONV; ++k)
      acc = fmaf(xsT[(t + k) * 84 + d], conv_w[d * DCONV + k], acc);
    xcT[t * 84 + d] = silu_f(acc);
  }
  __syncthreads();

  for (int idx = tid; idx < TL * XP; idx += blockDim.x) {
    int t = idx / XP, j = idx % XP;
    float acc = 0.f;
    for (int d = 0; d < DI; ++d)
      acc = fmaf(xcT[t * 84 + d], x_proj_w[j * DI + d], acc);
    xdT[t * 36 + j] = acc;
  }
  __syncthreads();

  for (int idx = tid; idx < TL * DI; idx += blockDim.x) {
    int t = idx / DI, d = idx % DI;
    float dv = dt_proj_b[d];
    dv = fmaf(xdT[t * 36 + 0], dt_proj_w[d * DTR + 0], dv);
    dv = fmaf(xdT[t * 36 + 1], dt_proj_w[d * DTR + 1], dv);
    dv = fmaf(xdT[t * 36 + 2], dt_proj_w[d * DTR + 2], dv);
    const long base = ((long)b * LL + l0 + t) * REC;
    rec[base + d]      = softplus_f(dv);
    rec[base + DI + d] = xcT[t * 84 + d];
  }
  for (int idx = tid; idx < TL * 32; idx += blockDim.x) {
    int t = idx >> 5, s = idx & 31;         // s 0..31 -> xdbc[3..34] = [B | C]
    const long base = ((long)b * LL + l0 + t) * REC;
    rec[base + 164 + s] = xdT[t * 36 + 3 + s];
  }
}

// ---------------- k3a: per-chunk scan with h0=0 -> (hF, P) ------------------
// Linear diagonal recurrence => chunk-parallel. 2 threads/channel x 8 states.
__global__ void k3a_chunk(const float* __restrict__ rec,
                          const float* __restrict__ A_log,
                          float* __restrict__ hF, float* __restrict__ Pb) {
  __shared__ __align__(16) float recT[TT * REC];
  const int b = blockIdx.x / NC, c = blockIdx.x % NC;
  const int tid = threadIdx.x;
  const int d = tid >> 1, hh = tid & 1;
  const bool act = (tid < 2 * DI);
  float h[8], P[8], A2[8];
  if (act) {
#pragma unroll
    for (int i = 0; i < 8; ++i) {
      A2[i] = -__expf(A_log[d * DSTATE + hh * 8 + i]) * 1.4426950408889634f;
      h[i] = 0.f; P[i] = 1.f;
    }
  }
  const long cbase = (long)b * LL + (long)c * CL;
  for (int lt = 0; lt < CL / TT; ++lt) {
    const long lbase = cbase + (long)lt * TT;
    const float* gsrc = rec + lbase * REC;          // 784-byte rows: 16B aligned
    for (int q = tid; q < (TT * REC) / 4; q += blockDim.x)
      async_g2l_b128(gsrc + q * 4, &recT[q * 4]);
    async_wait0();
    __syncthreads();
    if (act) {
      for (int t = 0; t < TT; ++t) {
        const float* r = &recT[t * REC];
        const float delta = r[d];
        const float dx    = delta * r[DI + d];
#pragma unroll
        for (int i = 0; i < 8; ++i) {
          const float dA = exp2f(delta * A2[i]);
          h[i] = fmaf(dA, h[i], dx * r[164 + hh * 8 + i]);
          P[i] *= dA;
        }
      }
    }
    __syncthreads();
  }
  if (act) {
    const long sb = (((long)b * NC + c) * DI + d) * DSTATE + hh * 8;
#pragma unroll
    for (int i = 0; i < 8; ++i) { hF[sb + i] = h[i]; Pb[sb + i] = P[i]; }
  }
}

// ---------------- k3b: sequential combine over 32 chunks (tiny) -------------
__global__ void k3b_combine(const float* __restrict__ hF,
                            const float* __restrict__ Pb,
                            float* __restrict__ Hinit) {
  const int b = blockIdx.x;
  for (int st = threadIdx.x; st < DI * DSTATE; st += blockDim.x) {
    float H = 0.f;
    for (int c = 0; c < NC; ++c) {
      const long idx = ((long)b * NC + c) * (DI * DSTATE) + st;
      Hinit[idx] = H;
      H = fmaf(Pb[idx], H, hF[idx]);
    }
  }
}

// ---------------- k3c: re-scan chunks from true init state -> y -------------
__global__ void k3c_scan(const float* __restrict__ rec,
                         const float* __restrict__ xz,
                         const float* __restrict__ A_log,
                         const float* __restrict__ Dvec,
                         const float* __restrict__ Hinit,
                         float* __restrict__ yg) {
  __shared__ __align__(16) float recT[TT * REC];
  __shared__ float zT[TT * DI];
  const int b = blockIdx.x / NC, c = blockIdx.x % NC;
  const int tid = threadIdx.x;
  const int d = tid >> 1, hh = tid & 1;
  const bool act = (tid < 2 * DI);
  float h[8], A2[8];
  float Dd = 0.f;
  if (act) {
    Dd = Dvec[d];
    const long sb = (((long)b * NC + c) * DI + d) * DSTATE + hh * 8;
#pragma unroll
    for (int i = 0; i < 8; ++i) {
      A2[i] = -__expf(A_log[d * DSTATE + hh * 8 + i]) * 1.4426950408889634f;
      h[i] = Hinit[sb + i];
    }
  }
  const long cbase = (long)b * LL + (long)c * CL;
  for (int lt = 0; lt < CL / TT; ++lt) {
    const long lbase = cbase + (long)lt * TT;
    const float* gsrc = rec + lbase * REC;
    for (int q = tid; q < (TT * REC) / 4; q += blockDim.x)
      async_g2l_b128(gsrc + q * 4, &recT[q * 4]);
    for (int idx = tid; idx < TT * DI; idx += blockDim.x)
      zT[idx] = xz[(lbase + idx / DI) * XZS + DI + (idx % DI)];
    async_wait0();
    __syncthreads();
    if (act) {
      for (int t = 0; t < TT; ++t) {
        const float* r = &recT[t * REC];
        const float delta = r[d];
        const float xcv   = r[DI + d];
        const float dx    = delta * xcv;
        float yacc = 0.f;
#pragma unroll
        for (int i = 0; i < 8; ++i) {
          const int s = hh * 8 + i;
          const float dA = exp2f(delta * A2[i]);
          h[i] = fmaf(dA, h[i], dx * r[164 + s]);
          yacc = fmaf(h[i], r[180 + s], yacc);
        }
        const float yo = yacc + __shfl_xor(yacc, 1, 32);
        if (hh == 0)
          yg[(lbase + t) * YS + d] = (yo + Dd * xcv) * silu_f(zT[t * DI + d]);
      }
    } else if (tid >= 164 && tid < 164 + 2 * TT) {
      const int t = (tid - 164) >> 1, p = (tid - 164) & 1;
      yg[(lbase + t) * YS + DI + p] = 0.f;  // zero K-pad lanes for k4
    }
    __syncthreads();
  }
}

// ---------------- k4: logits = y (M x 82) @ Wc^T + cls_b via fp32 WMMA ------
__global__ void k4_logits(const float* __restrict__ yg,
                          const float* __restrict__ WcP2,
                          const float* __restrict__ cls_b,
                          float* __restrict__ out) {
  const int lane = threadIdx.x & 31;
  const int wid  = blockIdx.x * (blockDim.x >> 5) + (threadIdx.x >> 5);
  if (wid >= MTOT / 16) return;
  const int l16 = lane & 15, hi = lane >> 4;
  const long m_a = (long)wid * 16 + l16;
  const float* ap = yg + m_a * YS + hi * 2;
  v8f c = {};
#pragma unroll
  for (int ks = 0; ks < YS / 4; ++ks) {     // 21 unconditional K-steps
    v2f av = *(const v2f*)(ap + ks * 4);
    const int kp = ks * 2 + hi;
    v2f bv = *(const v2f*)(WcP2 + (kp * 16 + l16) * 2);
    c = __builtin_amdgcn_wmma_f32_16x16x4_f32(false, av, false, bv,
                                              (short)0, c, false, false);
  }
#pragma unroll
  for (int r = 0; r < 8; ++r) {
    const long m = (long)wid * 16 + r + hi * 8;
    const int  n = l16;
    if (n < NL) out[m * NL + n] = c[r] + cls_b[n];
  }
}

// ---------------- launcher ----------------
extern "C" void kernel_launch(void* const* d_in, const int* in_sizes, int n_in,
                              void* d_out, int out_size, void* d_ws, size_t ws_size,
                              hipStream_t stream) {
  (void)in_sizes; (void)n_in; (void)out_size; (void)ws_size;
  const float* x          = (const float*)d_in[0];
  const float* in_proj_w  = (const float*)d_in[1];
  const float* conv_w     = (const float*)d_in[2];
  const float* conv_b     = (const float*)d_in[3];
  const float* x_proj_w   = (const float*)d_in[4];
  const float* dt_proj_w  = (const float*)d_in[5];
  const float* dt_proj_b  = (const float*)d_in[6];
  const float* A_log      = (const float*)d_in[7];
  const float* Dvec       = (const float*)d_in[8];
  const float* out_proj_w = (const float*)d_in[9];
  const float* cls_w      = (const float*)d_in[10];
  const float* cls_b      = (const float*)d_in[11];
  float* out = (float*)d_out;

  char* ws = (char*)d_ws;
  size_t off = 0;
  float* xP   = (float*)(ws + off); off += (size_t)MTOT * KXP * 4;   //  46 MB
  float* xz   = (float*)(ws + off); off += (size_t)MTOT * XZS * 4;   // 185 MB
  float* rec  = (float*)(ws + off); off += (size_t)MTOT * REC * 4;   // 206 MB
  float* yg   = (float*)(ws + off); off += (size_t)MTOT * YS  * 4;   //  88 MB
  float* hF   = (float*)(ws + off); off += (size_t)BB * NC * DI * DSTATE * 4;
  float* Pb   = (float*)(ws + off); off += (size_t)BB * NC * DI * DSTATE * 4;
  float* Hin  = (float*)(ws + off); off += (size_t)BB * NC * DI * DSTATE * 4;
  float* WinP = (float*)(ws + off); off += (size_t)11 * 22 * 16 * 2 * 4;
  float* WcP2 = (float*)(ws + off); off += (size_t)42 * 16 * 2 * 4;

  {
    const long n = (long)MTOT * KXP;
    kp_xpad<<<(int)((n + 255) / 256), 256, 0, stream>>>(x, xP);
  }
  kp_win<<<1, 256, 0, stream>>>(in_proj_w, WinP);
  k0_wc<<<1, 256, 0, stream>>>(cls_w, out_proj_w, WcP2);

  k1_inproj<<<(MTOT / 16) / 8, 256, 0, stream>>>(xP, WinP, xz);

  k2_feat<<<BB * (LL / TL), 256, 0, stream>>>(xz, conv_w, conv_b, x_proj_w,
                                              dt_proj_w, dt_proj_b, rec);

  k3a_chunk<<<BB * NC, 256, 0, stream>>>(rec, A_log, hF, Pb);
  k3b_combine<<<BB, 512, 0, stream>>>(hF, Pb, Hin);
  k3c_scan<<<BB * NC, 256, 0, stream>>>(rec, xz, A_log, Dvec, Hin, yg);

  k4_logits<<<(MTOT / 16) / 8, 256, 0, stream>>>(yg, WcP2, cls_b, out);
}